// RelationBlock_70300024701131
// MI455X (gfx1250) — compile-verified
//
#include <hip/hip_runtime.h>
#include <hip/hip_bf16.h>

// Problem constants from the reference
#define BB 8
#define NN 2048
#define DD 64

typedef __attribute__((ext_vector_type(2))) float v2f;
typedef __attribute__((ext_vector_type(4))) float v4f;
typedef __attribute__((ext_vector_type(8))) float v8f;
typedef __attribute__((ext_vector_type(4))) int   v4i;
typedef __attribute__((address_space(1))) v4i gv4i;   // global int4
typedef __attribute__((address_space(3))) v4i lv4i;   // LDS int4

#if __has_builtin(__builtin_amdgcn_global_load_async_to_lds_b128) && \
    __has_builtin(__builtin_amdgcn_s_wait_asynccnt)
#define USE_ASYNC_LDS 1
#else
#define USE_ASYNC_LDS 0
#endif

// ---------------------------------------------------------------------------
// Kernel 1: projections  q = x@Wq ; [k|v] = x@Wkv
// One block = 16 rows of x staged in LDS; 256 threads produce 16x192 outputs.
// ---------------------------------------------------------------------------
__global__ __launch_bounds__(256) void proj_kernel(
    const float* __restrict__ x, const float* __restrict__ Wq,
    const float* __restrict__ Wkv,
    float* __restrict__ q, float* __restrict__ k, float* __restrict__ v) {
  __shared__ float xs[16][DD];
  const int tid = threadIdx.x;
  const int rowbase = blockIdx.x * 16;

  for (int i = tid; i < 16 * DD; i += 256)
    xs[i >> 6][i & 63] = x[(size_t)(rowbase + (i >> 6)) * DD + (i & 63)];
  __syncthreads();

  const int r  = tid >> 4;   // 0..15 row within tile
  const int c0 = tid & 15;   // 0..15 starting output column
  const float* xr = xs[r];
  const size_t orow = (size_t)(rowbase + r) * DD;

  for (int o = c0; o < 192; o += 16) {
    float acc = 0.f;
    if (o < DD) {
#pragma unroll 8
      for (int t = 0; t < DD; ++t) acc += xr[t] * Wq[t * DD + o];
      q[orow + o] = acc;
    } else {
      const int f = o - DD;  // 0..127 column of Wkv
#pragma unroll 8
      for (int t = 0; t < DD; ++t) acc += xr[t] * Wkv[t * 2 * DD + f];
      if (f < DD) k[orow + f] = acc;
      else        v[orow + (f - DD)] = acc;
    }
  }
}

// ---------------------------------------------------------------------------
// Kernel 2: per-row argmax over q.k^T via V_WMMA_F32_16X16X4_F32 with k-tiles
// double-buffered in LDS (async global->LDS copy), then the fused
// retrieval + LN + FFN (pair@W folded to norm@(W1+W3) + x@W2) + LN + x.
// Block = 256 threads = 8 waves; each wave owns a 16-row q-tile; all waves
// share each staged k-tile (8x dedup of L2 traffic).
// ---------------------------------------------------------------------------
__global__ __launch_bounds__(256) void relation_kernel(
    const float* __restrict__ x,
    const float* __restrict__ q, const float* __restrict__ k,
    const float* __restrict__ v,
    const float* __restrict__ ln_mid_g, const float* __restrict__ ln_mid_b,
    const float* __restrict__ rn_in_W,  const float* __restrict__ rn_in_b,
    const float* __restrict__ rn_out_W, const float* __restrict__ rn_out_b,
    const float* __restrict__ ln_out_g, const float* __restrict__ ln_out_b,
    float* __restrict__ out) {
  // LDS-staged weights (L2-hot, loaded once per block)
  __shared__ float Wc[DD][DD];       // rn_in_W[0:64] + rn_in_W[128:192] (norm path)
  __shared__ float W2s[DD][DD];      // rn_in_W[64:128]                  (x path)
  __shared__ float Wo[DD][DD];       // rn_out_W
  __shared__ float vec[6][DD];       // ln_mid_g,ln_mid_b,rn_in_b,rn_out_b,ln_out_g,ln_out_b
  __shared__ int   jstar[128];
  __shared__ float ktile[2][16][DD]; // double-buffered k tile; epilogue scratch aliases this

  const int tid  = threadIdx.x;
  const int wave = tid >> 5;
  const int lane = tid & 31;

  // stage weights
  for (int i = tid; i < DD * DD; i += 256) {
    const int t = i >> 6, f2 = i & 63;
    Wc[t][f2]  = rn_in_W[t * DD + f2] + rn_in_W[(128 + t) * DD + f2];
    W2s[t][f2] = rn_in_W[(64 + t) * DD + f2];
    Wo[t][f2]  = rn_out_W[t * DD + f2];
  }
  for (int i = tid; i < DD; i += 256) {
    vec[0][i] = ln_mid_g[i]; vec[1][i] = ln_mid_b[i];
    vec[2][i] = rn_in_b[i];  vec[3][i] = rn_out_b[i];
    vec[4][i] = ln_out_g[i]; vec[5][i] = ln_out_b[i];
  }

  const int b        = blockIdx.x >> 4;        // batch
  const int tileBase = (blockIdx.x & 15) * 128;
  const int ibase    = tileBase + wave * 16;   // first q row (within batch)

  const float* qb = q + (size_t)b * NN * DD;
  const float* kb = k + (size_t)b * NN * DD;
  const float* vb = v + (size_t)b * NN * DD;
  const float* xb = x + (size_t)b * NN * DD;

  // block-wide cooperative k-tile fetch: thread copies one 16B segment
  const int crow = tid >> 4;   // 0..15 row of the k tile
  const int cseg = tid & 15;   // 0..15 16-byte segment within row
  auto fetch = [&](int jt, int bufIdx) {
    const float* src = kb + (size_t)(jt * 16 + crow) * DD + cseg * 4;
    float* dst = &ktile[bufIdx][crow][cseg * 4];
#if USE_ASYNC_LDS
    __builtin_amdgcn_global_load_async_to_lds_b128((gv4i*)src, (lv4i*)dst, 0, 0);
#else
    v4f tmp = *(const v4f*)src;
    *(v4f*)dst = tmp;
#endif
  };

  // ---- A operands: q tile, 16 K-chunks of 2 VGPRs (ISA 16x4 f32 layout) ----
  const int mrow  = lane & 15;          // M (and N for B operand)
  const int kpair = (lane >> 4) << 1;   // K offset 0 or 2 within chunk
  v2f a[16];
  {
    const float* qrow = qb + (size_t)(ibase + mrow) * DD + kpair;
#pragma unroll
    for (int kbk = 0; kbk < 16; ++kbk)
      a[kbk] = *(const v2f*)(qrow + kbk * 4);
  }

  float best[8];
  int   bestj[8];
#pragma unroll
  for (int r = 0; r < 8; ++r) { best[r] = -3.4e38f; bestj[r] = 0; }

  // prologue: stage tile 0
  fetch(0, 0);

  // ---- sweep all 128 j-tiles --------------------------------------------
  for (int jt = 0; jt < NN / 16; ++jt) {
    const int buf = jt & 1;
#if USE_ASYNC_LDS
    __builtin_amdgcn_s_wait_asynccnt(0);   // this wave's slice of tile jt landed
#endif
    __syncthreads();  // tile jt visible; all waves done reading buf^1 (iter jt-1)
    if (jt + 1 < NN / 16) fetch(jt + 1, buf ^ 1);  // overlaps with compute below

    const float* krow = &ktile[buf][mrow][kpair];
    v8f c = {};
#pragma unroll
    for (int kbk = 0; kbk < 16; ++kbk) {
      v2f bvec = *(const v2f*)(krow + kbk * 4);   // ds_load_b64, conflict-free
      c = __builtin_amdgcn_wmma_f32_16x16x4_f32(
          false, a[kbk], false, bvec, (short)0, c, false, false);
    }
    const int j = jt * 16 + mrow;  // this lane's N column
#pragma unroll
    for (int r = 0; r < 8; ++r) {
      const float val = c[r];
      if (val > best[r]) { best[r] = val; bestj[r] = j; }  // j monotone => first-max kept
    }
  }

  // ---- reduce (max, argmax) across the 16-lane half holding each row -------
#pragma unroll
  for (int r = 0; r < 8; ++r) {
#pragma unroll
    for (int off = 1; off < 16; off <<= 1) {
      const float ov = __shfl_xor(best[r], off, 32);
      const int   oj = __shfl_xor(bestj[r], off, 32);
      if (ov > best[r] || (ov == best[r] && oj < bestj[r])) {
        best[r] = ov; bestj[r] = oj;
      }
    }
  }
  if ((lane & 15) == 0) {
    const int rb = wave * 16 + ((lane >> 4) << 3);  // rows 0..7 or 8..15
#pragma unroll
    for (int r = 0; r < 8; ++r) jstar[rb + r] = bestj[r];
  }
  __syncthreads();  // jstar ready; k tiles dead -> reuse ktile as scratch

  // epilogue scratch aliases the (now dead) k-tile buffer: 3 x 8 x 64 floats
  float* scratch = &ktile[0][0][0];
  float* nrmw = scratch + wave * DD;            // norm row of this wave
  float* xsrw = scratch + 8 * DD + wave * DD;   // x row of this wave
  float* midw = scratch + 16 * DD + wave * DD;  // mid row of this wave

  // ---- epilogue: per row, gather v[j*], LN, FFN, LN, +x --------------------
  const int f = lane * 2;  // each lane owns features f, f+1
  for (int r = 0; r < 16; ++r) {
    const int i  = ibase + r;                 // row within batch
    const int js = jstar[wave * 16 + r];

    v2f vv = *(const v2f*)(vb + (size_t)js * DD + f);
    v2f xv = *(const v2f*)(xb + (size_t)i  * DD + f);

    // LayerNorm(mid) stats over 64 values (2 per lane)
    float s1 = vv.x + vv.y;
    float s2 = vv.x * vv.x + vv.y * vv.y;
#pragma unroll
    for (int off = 16; off >= 1; off >>= 1) {
      s1 += __shfl_xor(s1, off, 32);
      s2 += __shfl_xor(s2, off, 32);
    }
    const float m   = s1 * (1.f / DD);
    const float inv = rsqrtf(s2 * (1.f / DD) - m * m + 1e-5f);
    nrmw[f]     = (vv.x - m) * inv * vec[0][f]     + vec[1][f];
    nrmw[f + 1] = (vv.y - m) * inv * vec[0][f + 1] + vec[1][f + 1];
    xsrw[f]     = xv.x;
    xsrw[f + 1] = xv.y;
    __syncthreads();

    // mid = relu(norm@(W1+W3) + x@W2 + b_in)
    float a0 = vec[2][f], a1 = vec[2][f + 1];
#pragma unroll 8
    for (int t = 0; t < DD; ++t) {
      const float nt = nrmw[t], xt = xsrw[t];
      a0 += nt * Wc[t][f]     + xt * W2s[t][f];
      a1 += nt * Wc[t][f + 1] + xt * W2s[t][f + 1];
    }
    midw[f]     = fmaxf(a0, 0.f);
    midw[f + 1] = fmaxf(a1, 0.f);
    __syncthreads();

    // rn = mid@rn_out_W + b_out
    float r0 = vec[3][f], r1 = vec[3][f + 1];
#pragma unroll 8
    for (int t = 0; t < DD; ++t) {
      const float mt = midw[t];
      r0 += mt * Wo[t][f];
      r1 += mt * Wo[t][f + 1];
    }

    // out = LN(rn)*g+b + x
    float t1 = r0 + r1;
    float t2 = r0 * r0 + r1 * r1;
#pragma unroll
    for (int off = 16; off >= 1; off >>= 1) {
      t1 += __shfl_xor(t1, off, 32);
      t2 += __shfl_xor(t2, off, 32);
    }
    const float m2   = t1 * (1.f / DD);
    const float inv2 = rsqrtf(t2 * (1.f / DD) - m2 * m2 + 1e-5f);
    const size_t orow = (size_t)b * NN * DD + (size_t)i * DD;
    out[orow + f]     = (r0 - m2) * inv2 * vec[4][f]     + vec[5][f]     + xsrw[f];
    out[orow + f + 1] = (r1 - m2) * inv2 * vec[4][f + 1] + vec[5][f + 1] + xsrw[f + 1];
    __syncthreads();
  }
}

// ---------------------------------------------------------------------------
extern "C" void kernel_launch(void* const* d_in, const int* in_sizes, int n_in,
                              void* d_out, int out_size, void* d_ws, size_t ws_size,
                              hipStream_t stream) {
  const float* x        = (const float*)d_in[0];
  const float* Wq       = (const float*)d_in[1];
  const float* Wkv      = (const float*)d_in[2];
  const float* ln_mid_g = (const float*)d_in[3];
  const float* ln_mid_b = (const float*)d_in[4];
  const float* rn_in_W  = (const float*)d_in[5];
  const float* rn_in_b  = (const float*)d_in[6];
  const float* rn_out_W = (const float*)d_in[7];
  const float* rn_out_b = (const float*)d_in[8];
  const float* ln_out_g = (const float*)d_in[9];
  const float* ln_out_b = (const float*)d_in[10];
  float* out = (float*)d_out;

  const size_t BND = (size_t)BB * NN * DD;  // 1M floats = 4MB
  float* q = (float*)d_ws;
  float* k = q + BND;
  float* v = k + BND;

  proj_kernel<<<(BB * NN) / 16, 256, 0, stream>>>(x, Wq, Wkv, q, k, v);
  relation_kernel<<<BB * (NN / 128), 256, 0, stream>>>(
      x, q, k, v, ln_mid_g, ln_mid_b, rn_in_W, rn_in_b, rn_out_W, rn_out_b,
      ln_out_g, ln_out_b, out);
}